// PolyaTree_64132451664657
// MI455X (gfx1250) — compile-verified
//
#include <hip/hip_runtime.h>

typedef __attribute__((ext_vector_type(2))) float v2f;
typedef __attribute__((ext_vector_type(8))) float v8f;

#define PT_DIM    16
#define PT_L      8
#define PT_NODES  255   // 2^8 - 1

// Walk one dim's tree exactly as the reference builds/reads its interval
// tables. `fp contract(off)` is essential: split/rhi must be the separately
// rounded mul+add the reference computes, or 1-ulp fma differences flip
// left/right decisions on a handful of the 112M comparisons.
__device__ __forceinline__ float pt_traverse(const float* __restrict__ row, float xv) {
#pragma clang fp contract(off)
  float lo = 0.0f, hi = 1.0f, prodb = 1.0f;
  int node = 0;
#pragma unroll
  for (int l = 0; l < PT_L - 1; ++l) {
    float b     = row[node];                  // ds_load_b32 gather from LDS
    prodb       = prodb * b;
    float len   = hi - lo;
    float split = lo + b * len;               // no fma: bit-matches reference
    float rhi   = split + (1.0f - b) * len;   // no fma: right-child upper bound
    bool left   = (xv <= split);
    hi   = left ? split : rhi;
    lo   = left ? lo    : split;
    node = left ? (2 * node + 1) : (2 * node + 2);
  }
  float b = row[node];                        // level L-1: take beta, no descend
  prodb   = prodb * b;
  float leaf = hi - lo;
  // sum(log b_l) - log(leaf) == log(prod b / leaf): 1 logf + v_rcp_f32
  // instead of 9 logf; <=2 ulp relative error -> ~1e-7 absolute in the log.
  return logf(prodb * __builtin_amdgcn_rcpf(leaf));
}

__global__ __launch_bounds__(256)
void polya_tree_kernel(const float* __restrict__ x,
                       const float* __restrict__ samples,
                       float* __restrict__ out, int n) {
  __shared__ float sb[PT_DIM * PT_NODES];      // 16320 B beta table

  const int tid = threadIdx.x;
  for (int i = tid; i < PT_DIM * PT_NODES; i += 256) sb[i] = samples[i];

  const int wave = tid >> 5;        // 8 waves / block, 16 samples / wave
  const int lane = tid & 31;
  const int half = lane >> 4;       // which half-wave (A-matrix K-group)
  const int m    = lane & 15;       // sample row within the wave's 16x16 tile
  const int wavebase = blockIdx.x * 128 + wave * 16;

  const int s0 = wavebase + m;
  const int sc = (s0 < n) ? s0 : (n - 1);      // clamp, never diverge pre-WMMA
  const float* xr = x + (size_t)sc * PT_DIM;
  __builtin_prefetch(xr, 0, 0);                // global_prefetch_b8 while LDS fills

  __syncthreads();

  // Lane's 8 dims: chunk c covers dims 4c..4c+3; this half takes 2 of them.
  v2f xa[4];
#pragma unroll
  for (int c = 0; c < 4; ++c)
    xa[c] = *(const v2f*)(xr + 4 * c + 2 * half);   // 8B-aligned float2

#if __has_builtin(__builtin_amdgcn_wmma_f32_16x16x4_f32)
  // A (16x4 f32): lane m holds K=2*half{,+1} in its 2 VGPRs -> row m = sample m,
  // K 0..3 = dims 4c..4c+3. B = ones => D[m][*] = sum_k A[m][k]; accumulate
  // over the 4 chunks to reduce all 16 dims in fp32-exact WMMA math.
  v8f acc = {0.f, 0.f, 0.f, 0.f, 0.f, 0.f, 0.f, 0.f};
#pragma unroll
  for (int c = 0; c < 4; ++c) {
    const int d0 = 4 * c + 2 * half;
    float r0 = pt_traverse(sb + (size_t)d0 * PT_NODES,       xa[c].x);
    float r1 = pt_traverse(sb + (size_t)(d0 + 1) * PT_NODES, xa[c].y);
    v2f a;  a.x = r0;  a.y = r1;
    v2f bo; bo.x = 1.0f; bo.y = 1.0f;
    acc = __builtin_amdgcn_wmma_f32_16x16x4_f32(
        /*neg_a=*/false, a, /*neg_b=*/false, bo,
        /*c_mod=*/(short)0, acc, /*reuse_a=*/false, /*reuse_b=*/false);
  }
  // D layout: VGPR j, lanes 0-15 -> M=j, lanes 16-31 -> M=j+8 (col N = m).
  // Lane (half, m<8) emits sample M = 8*half + m from acc[m].
  if (m < 8) {
    float v = (m == 0) ? acc[0] : (m == 1) ? acc[1] : (m == 2) ? acc[2] :
              (m == 3) ? acc[3] : (m == 4) ? acc[4] : (m == 5) ? acc[5] :
              (m == 6) ? acc[6] : acc[7];
    int sdst = wavebase + 8 * half + m;
    if (sdst < n) out[sdst] = v * 0.0625f;     // mean over 16 dims
  }
#else
  // Fallback: lane-local partial + cross-half shuffle reduce.
  float part = 0.0f;
#pragma unroll
  for (int c = 0; c < 4; ++c) {
    const int d0 = 4 * c + 2 * half;
    part += pt_traverse(sb + (size_t)d0 * PT_NODES,       xa[c].x);
    part += pt_traverse(sb + (size_t)(d0 + 1) * PT_NODES, xa[c].y);
  }
  part += __shfl_xor(part, 16, 32);
  if (half == 0 && s0 < n) out[s0] = part * 0.0625f;
#endif
}

extern "C" void kernel_launch(void* const* d_in, const int* in_sizes, int n_in,
                              void* d_out, int out_size, void* d_ws, size_t ws_size,
                              hipStream_t stream) {
  const float* x       = (const float*)d_in[0];   // (n, 16) f32
  const float* samples = (const float*)d_in[1];   // (16, 255) f32
  (void)d_ws; (void)ws_size; (void)n_in;          // L (d_in[2]) fixed at 8
  float* out = (float*)d_out;                     // (n,) f32
  const int n = in_sizes[0] / PT_DIM;
  const int blocks = (n + 127) / 128;             // 128 samples per 256-thread block
  polya_tree_kernel<<<blocks, 256, 0, stream>>>(x, samples, out, n);
}